// JaggedMaxModule_30150670418631
// MI455X (gfx1250) — compile-verified
//
#include <hip/hip_runtime.h>
#include <hip/hip_bf16.h>
#include <cstdint>

// Jagged segment-max, MI455X (gfx1250).
//
// values:  [total, 128] fp32   (~1.05 GB streamed once)
// prefix:  [n_seg] int32 cumulative ends
// out:     [n_seg, 128] fp32 per-segment max
//
// Pure bandwidth problem (~1.10 GB traffic -> ~47us floor @ 23.3 TB/s).
// One wave32 per segment; lane l owns a float4 of features 4l..4l+3, so each
// row is one fully-coalesced 512B burst. Rows are staged global->LDS with the
// CDNA5 async copy engine (global_load_async_to_lds_b128 / ASYNCcnt) in a
// double-buffered pipeline, then reduced out of LDS with ds_load_b128 +
// v_max_num_f32. Padded stage slots re-load the segment's LAST row (valid
// data, cache-hit), so every stage reduces exactly CHUNK rows with no
// data-dependent inner loop.

#define FEAT            128
#define WAVES_PER_BLOCK 8
#define CHUNK           4   // rows per async stage; each row = 512B

__device__ __forceinline__ void fmax4(float4& a, const float4 b) {
    a.x = __builtin_fmaxf(a.x, b.x);
    a.y = __builtin_fmaxf(a.y, b.y);
    a.z = __builtin_fmaxf(a.z, b.z);
    a.w = __builtin_fmaxf(a.w, b.w);
}

__global__ void __launch_bounds__(WAVES_PER_BLOCK * 32)
jagged_segmax_kernel(const float* __restrict__ values,
                     const int*   __restrict__ prefix,
                     float*       __restrict__ out,
                     int n_seg) {
    // [wave][buf][row][lane] float4 = 8 * 2 * 4 * 32 * 16B = 32 KB
    __shared__ float4 smem[WAVES_PER_BLOCK * 2 * CHUNK * 32];

    const int lane = threadIdx.x & 31;
    const int wave = threadIdx.x >> 5;
    const int seg  = blockIdx.x * WAVES_PER_BLOCK + wave;
    if (seg >= n_seg) return;   // no block-wide barriers used; safe to diverge

    const int start = (seg == 0) ? 0 : prefix[seg - 1];
    const int end   = prefix[seg];                 // exclusive
    const int rows  = end - start;

    float4* wbuf = &smem[wave * (2 * CHUNK * 32)];
    // Low 32 bits of a shared-aperture generic address are the wave-relative
    // LDS byte address (ISA 10.2) — exactly what the async VDST operand wants.
    const unsigned lds_base = (unsigned)(uintptr_t)wbuf;
    const uint64_t vbase    = (uint64_t)(uintptr_t)values;
    const int      last_row = end - 1;

    // Issue one stage: always exactly CHUNK async b128 copies (512B/row for
    // the whole wave), clamping the row index so padded slots re-read the
    // segment's last row (in-bounds, valid for max, L2-resident).
    auto issue = [&](int buf, int base_row) {
        // Our earlier LDS reads of this region must complete before the async
        // engine starts overwriting it.
        asm volatile("s_wait_dscnt 0x0" ::: "memory");
#pragma unroll
        for (int r = 0; r < CHUNK; ++r) {
            int row = base_row + r;
            row = (row < last_row) ? row : last_row;
            uint64_t g = vbase +
                (((uint64_t)row * FEAT + (uint64_t)(lane * 4)) * sizeof(float));
            unsigned l = lds_base +
                (unsigned)(((buf * CHUNK + r) * 32 + lane) * sizeof(float4));
            asm volatile("global_load_async_to_lds_b128 %0, %1, off"
                         :: "v"(l), "v"(g)
                         : "memory");
        }
    };

    const int stages = (rows + CHUNK - 1) / CHUNK;

    float4 acc;
    acc.x = acc.y = acc.z = acc.w = -__builtin_inff();

    issue(0, start);          // prologue: fill buffer 0
    int buf = 0;
    for (int s = 0; s < stages; ++s) {
        if (s + 1 < stages) {
            issue(buf ^ 1, start + (s + 1) * CHUNK);  // prefetch next stage
            // The CHUNK ops just issued may still be in flight; anything
            // older (this stage's buffer) is guaranteed complete (in-order).
            asm volatile("s_wait_asynccnt 0x4" ::: "memory");
        } else {
            asm volatile("s_wait_asynccnt 0x0" ::: "memory");
        }

        // Unconditionally reduce all CHUNK rows: padded slots hold duplicates
        // of the segment's last row, which max() absorbs.
        const float4* bp = &wbuf[buf * (CHUNK * 32) + lane];
#pragma unroll
        for (int r = 0; r < CHUNK; ++r) {
            fmax4(acc, bp[r * 32]);                   // ds_load_b128 + v_max
        }
        buf ^= 1;
    }

    float4* o = reinterpret_cast<float4*>(out);
    o[(size_t)seg * 32 + lane] = acc;                 // global_store_b128
}

extern "C" void kernel_launch(void* const* d_in, const int* in_sizes, int n_in,
                              void* d_out, int out_size, void* d_ws, size_t ws_size,
                              hipStream_t stream) {
    const float* values = (const float*)d_in[0];
    const int*   prefix = (const int*)d_in[1];
    float*       out    = (float*)d_out;

    const int n_seg  = in_sizes[1];
    const int blocks = (n_seg + WAVES_PER_BLOCK - 1) / WAVES_PER_BLOCK;

    jagged_segmax_kernel<<<dim3(blocks), dim3(WAVES_PER_BLOCK * 32), 0, stream>>>(
        values, prefix, out, n_seg);
}